// RawMomentsLayer_86586540687592
// MI455X (gfx1250) — compile-verified
//
#include <hip/hip_runtime.h>
#include <stdint.h>

typedef __attribute__((ext_vector_type(2))) float v2f;
typedef __attribute__((ext_vector_type(8))) float v8f;

#define TILE_B      16     // batches per block (WMMA M)
#define KC          256    // k-chunk per iteration (2 image rows)
#define LDS_STRIDE  260    // 256 + 4 pad -> conflict-free ds_load_b64 fragments
#define TILE_FLOATS (TILE_B * LDS_STRIDE)   // 4160 floats = 16640 B per buffer
#define NWAVES      8      // waves per block (wave32)
#define IMG_ELEMS   16384  // 128*128
#define NITER       (IMG_ELEMS / KC)        // 64

// out[b, j] = sum_k img[b, k] * W[k, j],  W[k=(y*128+x), j] = x^p(j) * y^q(j)
// j order: m00 m10 m01 m20 m11 m02 m30 m21 m12 m03
__global__ __launch_bounds__(256)
void raw_moments_wmma(const float* __restrict__ img, float* __restrict__ out,
                      int batch)
{
    __shared__ float tile[2 * TILE_FLOATS];   // double-buffered A tiles
    __shared__ float red[NWAVES * 256];

    const int t    = threadIdx.x;
    const int lane = t & 31;
    const int w    = t >> 5;          // wave id 0..7
    const int b0   = blockIdx.x * TILE_B;

    const int n  = lane & 15;         // output column (N) this lane feeds in B
    const int hi = lane >> 4;         // 0: K pair {0,1}, 1: K pair {2,3}

    // (p,q) exponents per output column; columns 10..15 are dead (zero weight)
    static const int Ptab[16] = {0,1,0,2,1,0,3,2,1,0, 0,0,0,0,0,0};
    static const int Qtab[16] = {0,0,1,0,1,2,0,1,2,3, 0,0,0,0,0,0};
    const int  p    = Ptab[n];
    const int  q    = Qtab[n];
    const bool live = (n < 10);

    // This wave's fixed x-range: x(s,v) = (w&3)*32 + s*4 + 2*hi + v.
    float xp[8][2];
#pragma unroll
    for (int s = 0; s < 8; ++s) {
#pragma unroll
        for (int v = 0; v < 2; ++v) {
            float x = (float)((w & 3) * 32 + s * 4 + 2 * hi + v);
            float val = 1.0f;
            if (p >= 1) val *= x;
            if (p >= 2) val *= x;
            if (p >= 3) val *= x;
            xp[s][v] = live ? val : 0.0f;
        }
    }

    const int colbase = ((w >> 2) * 128) + ((w & 3) * 32);
    const int arow    = lane & 15;    // A-matrix M row for this lane

    // ---- Async DMA staging setup: 4x 16B per thread per tile --------------
    // Per-thread global addresses (advance 1024 B per k-chunk) and LDS byte
    // offsets (low 32 bits of the flat LDS pointer == LDS allocation offset).
    uint64_t gaddr[4];
    uint32_t laddr[4];
#pragma unroll
    for (int rr = 0; rr < 4; ++rr) {
        int lin = rr * 1024 + t * 4;       // linear float index in 16x256 tile
        int row = lin >> 8;                // 0..15
        int col = lin & 255;               // 0..255
        gaddr[rr] = (uint64_t)(uintptr_t)(img + (size_t)(b0 + row) * IMG_ELEMS + col);
        laddr[rr] = (uint32_t)(uintptr_t)(&tile[row * LDS_STRIDE + col]);
    }

    // Prologue: DMA chunk 0 into buffer 0.
#pragma unroll
    for (int rr = 0; rr < 4; ++rr) {
        asm volatile("global_load_async_to_lds_b128 %0, %1, off"
                     :: "v"(laddr[rr]), "v"(gaddr[rr]) : "memory");
        gaddr[rr] += (uint64_t)KC * sizeof(float);   // +1024 B
    }

    v8f acc = {0.f, 0.f, 0.f, 0.f, 0.f, 0.f, 0.f, 0.f};

    for (int it = 0; it < NITER; ++it) {
        // Current buffer's DMA complete (own wave), then block-wide visibility.
        asm volatile("s_wait_asynccnt 0x0" ::: "memory");
        __syncthreads();

        // Kick off DMA for the next chunk into the other buffer while we compute.
        if (it + 1 < NITER) {
            const uint32_t bufofs = (uint32_t)(((it + 1) & 1) * TILE_FLOATS * sizeof(float));
#pragma unroll
            for (int rr = 0; rr < 4; ++rr) {
                asm volatile("global_load_async_to_lds_b128 %0, %1, off"
                             :: "v"(laddr[rr] + bufofs), "v"(gaddr[rr]) : "memory");
                gaddr[rr] += (uint64_t)KC * sizeof(float);
            }
        }

        const float* tb = tile + (it & 1) * TILE_FLOATS;

        // y^q for this chunk (y fixed per wave within the chunk)
        float y  = (float)(it * 2 + (w >> 2));
        float yq = 1.0f;
        if (q >= 1) yq *= y;
        if (q >= 2) yq *= y;
        if (q >= 3) yq *= y;

        // 8 WMMA steps over this wave's 32-k slice.
#pragma unroll
        for (int s = 0; s < 8; ++s) {
            int c = colbase + s * 4 + 2 * hi;
            v2f a = *reinterpret_cast<const v2f*>(&tb[arow * LDS_STRIDE + c]);
            v2f b;
            b.x = xp[s][0] * yq;
            b.y = xp[s][1] * yq;
            acc = __builtin_amdgcn_wmma_f32_16x16x4_f32(
                /*neg_a=*/false, a, /*neg_b=*/false, b,
                /*c_mod=*/(short)0, acc, /*reuse_a=*/false, /*reuse_b=*/false);
        }
    }

    // Cross-wave reduction of the eight 16x16 partial tiles.
    __syncthreads();
#pragma unroll
    for (int r = 0; r < 8; ++r)
        red[w * 256 + r * 32 + lane] = acc[r];
    __syncthreads();

    float sum = 0.f;
#pragma unroll
    for (int ww = 0; ww < NWAVES; ++ww)
        sum += red[ww * 256 + t];

    // C/D layout: VGPR r, lane l -> M = r + 8*(l>>4), N = l&15.
    int r = t >> 5;
    int l = t & 31;
    int M = r + 8 * (l >> 4);
    int N = l & 15;
    if (N < 10 && (b0 + M) < batch)
        out[(size_t)(b0 + M) * 10 + N] = sum;
}

extern "C" void kernel_launch(void* const* d_in, const int* in_sizes, int n_in,
                              void* d_out, int out_size, void* d_ws, size_t ws_size,
                              hipStream_t stream) {
    (void)n_in; (void)out_size; (void)d_ws; (void)ws_size;
    const float* img = (const float*)d_in[0];
    float* out = (float*)d_out;

    int batch  = in_sizes[0] / IMG_ELEMS;       // 4096
    int blocks = (batch + TILE_B - 1) / TILE_B; // 256

    raw_moments_wmma<<<dim3(blocks), dim3(256), 0, stream>>>(img, out, batch);
}